// GCN_51135880626696
// MI455X (gfx1250) — compile-verified
//
#include <hip/hip_runtime.h>

typedef __attribute__((ext_vector_type(2))) float v2f;
typedef __attribute__((ext_vector_type(8))) float v8f;

#define N_NODES 50000
#define N_EDGES 800000
#define DIM     128

// ---------------------------------------------------------------------------
// Stage 0: zero the aggregation workspace (agg lives in d_ws).
// ---------------------------------------------------------------------------
__global__ void gcn_zero_kernel(float4* __restrict__ p, int n4) {
    int i = blockIdx.x * blockDim.x + threadIdx.x;
    if (i < n4) p[i] = make_float4(0.f, 0.f, 0.f, 0.f);
}

// ---------------------------------------------------------------------------
// Stage 1: edge scatter-aggregate. One wave32 per edge; each lane covers a
// float4 slice of the 128-dim feature row. x and agg are L2-resident
// (25.6 MB each vs 192 MB L2), so gathers + f32 atomics stay on-chip.
// ---------------------------------------------------------------------------
__global__ void __launch_bounds__(256)
gcn_scatter_kernel(const float* __restrict__ x,
                   const int*   __restrict__ esrc,
                   const int*   __restrict__ edst,
                   const float* __restrict__ ew,
                   float*       __restrict__ agg) {
    int tid  = blockIdx.x * blockDim.x + threadIdx.x;
    int e    = tid >> 5;          // edge index (wave-uniform)
    int lane = tid & 31;          // lane -> 4-float slice
    if (e >= N_EDGES) return;

    int   s  = esrc[e];           // wave-uniform -> scalar loads
    int   d  = edst[e];
    float wt = ew[e];

    const float4 v = ((const float4*)(x + (size_t)s * DIM))[lane];  // b128 load
    float* ap = agg + (size_t)d * DIM + lane * 4;
    atomicAdd(ap + 0, v.x * wt);
    atomicAdd(ap + 1, v.y * wt);
    atomicAdd(ap + 2, v.z * wt);
    atomicAdd(ap + 3, v.w * wt);
}

// ---------------------------------------------------------------------------
// Stage 2: out = agg @ W + b via V_WMMA_F32_16X16X4_F32 (f32 in/out, matches
// the reference precision). One wave per 16x16 output tile, K walked in steps
// of 4 (32 WMMA issues). Block = 8 waves = the 8 column tiles of one row
// tile; grid.x = 50000/16 = 3125 row tiles (exact -> EXEC all-1s for WMMA).
//
// Fragment maps (ISA 7.12.2, 32-bit layouts):
//   A 16x4 : lane m=L%16; lanes 0-15 hold K={0,1} in v[0..1], lanes 16-31
//            hold K={2,3}.
//   B 4x16 : lane n=L%16; same K split across lane halves.
//   C/D    : v[r] -> row M = r + 8*(L/16), col N = L%16.
// ---------------------------------------------------------------------------
__global__ void __launch_bounds__(256)
gcn_gemm_wmma_kernel(const float* __restrict__ agg,
                     const float* __restrict__ W,
                     const float* __restrict__ bias,
                     float*       __restrict__ out) {
    const int lane    = threadIdx.x & 31;
    const int wave    = threadIdx.x >> 5;   // 0..7 -> column tile
    const int rowTile = blockIdx.x;         // 0..3124
    const int mn      = lane & 15;          // A row (m) and B col (n)
    const int half    = lane >> 4;          // K sub-pair selector
    const int colBase = wave * 16;

    const float* arow = agg + (size_t)(rowTile * 16 + mn) * DIM;

    v8f c = {};
    #pragma unroll
    for (int k0 = 0; k0 < DIM; k0 += 4) {
        const int kb = k0 + half * 2;
        v2f a;
        a.x = arow[kb];
        a.y = arow[kb + 1];
        v2f b;
        b.x = W[(size_t)kb * DIM + colBase + mn];
        b.y = W[(size_t)(kb + 1) * DIM + colBase + mn];
        // 8 args: (neg_a, A, neg_b, B, c_mod, C, reuse_a, reuse_b)
        c = __builtin_amdgcn_wmma_f32_16x16x4_f32(
                false, a, false, b, (short)0, c, false, false);
    }

    const float bv    = bias[colBase + mn];
    const int   mBase = half * 8;
    #pragma unroll
    for (int r = 0; r < 8; ++r) {
        out[(size_t)(rowTile * 16 + mBase + r) * DIM + colBase + mn] = c[r] + bv;
    }
}

// ---------------------------------------------------------------------------
// Launch: zero agg -> edge scatter -> WMMA GEMM, all on `stream`.
// d_in order: x, edge_src, edge_dst, edge_weight, W, b.
// ---------------------------------------------------------------------------
extern "C" void kernel_launch(void* const* d_in, const int* in_sizes, int n_in,
                              void* d_out, int out_size, void* d_ws, size_t ws_size,
                              hipStream_t stream) {
    const float* x    = (const float*)d_in[0];
    const int*   esrc = (const int*)d_in[1];
    const int*   edst = (const int*)d_in[2];
    const float* ew   = (const float*)d_in[3];
    const float* W    = (const float*)d_in[4];
    const float* b    = (const float*)d_in[5];
    float*       out  = (float*)d_out;
    float*       agg  = (float*)d_ws;   // 50000*128*4 = 25.6 MB scratch

    // Zero agg (float4 stores).
    const int n4 = N_NODES * DIM / 4;
    gcn_zero_kernel<<<(n4 + 255) / 256, 256, 0, stream>>>((float4*)agg, n4);

    // Scatter: one wave per edge -> 800000*32 threads.
    const long long sthreads = (long long)N_EDGES * 32;
    gcn_scatter_kernel<<<(int)((sthreads + 255) / 256), 256, 0, stream>>>(
        x, esrc, edst, ew, agg);

    // GEMM: 3125 row tiles x (8 waves = 8 col tiles) per block.
    gcn_gemm_wmma_kernel<<<N_NODES / 16, 256, 0, stream>>>(agg, W, b, out);
}